// BoundaryLoss_70652212019209
// MI455X (gfx1250) — compile-verified
//
#include <hip/hip_runtime.h>
#include <math.h>
#include <stdint.h>

// ---------------------------------------------------------------------------
// BoundaryLoss for MI455X (gfx1250, wave32).
//
// Hot loop = 64 min-plus 9-pt stencil sweeps (not a matmul -> WMMA N/A).
// CDNA5 features used instead:
//   * 320KB-per-WGP LDS: 8 Jacobi iterations per global pass with an 8-px
//     redundant halo (exact equivalence to 64 full-image sweeps).
//   * Tensor Data Mover: one 2D-descriptor tensor_load_to_lds per interior
//     tile (TENSORcnt / s_wait_tensorcnt).
//   * Async global->LDS per-lane copies for border tiles (ASYNCcnt).
//   * global_prefetch_b8 in the elementwise pass; wave32-safe fixed-order
//     reductions (deterministic across graph replays).
// ---------------------------------------------------------------------------

#define BIGD   1e4f
#define W_EDGE 0.955f
#define W_DIAG 1.3693f

#define BATCH 16
#define IMH   512
#define IMW   512
#define NPIX  (BATCH * IMH * IMW)      // 4194304

#define TILE   64
#define RIT    8                       // LDS relaxation iterations per pass
#define HALO   RIT                     // halo >= iterations -> exact center
#define IDIM   (TILE + 2 * HALO)       // 80 : loaded interior
#define PDIM   (IDIM + 2)              // 82 : + BIG guard ring (= jnp pad)
#define PCELLS (PDIM * PDIM)           // 6724
#define ICELLS (IDIM * IDIM)           // 6400
#define NPASS  8                       // 8 passes * 8 iters = 64 sweeps

#define THREADS     256                // 8 wave32 waves per block
#define CPT         (ICELLS / THREADS) // 25 cells per thread
#define OPT         (TILE * TILE / THREADS) // 16 outputs per thread
#define LOSS_BLOCKS 1024
#define PER_THREAD  16                 // 1024*256*16 == NPIX

// -------- CDNA5 async global->LDS path (guarded) ----------------------------
#if defined(__has_builtin)
#if __has_builtin(__builtin_amdgcn_global_load_async_to_lds_b32) && \
    __has_builtin(__builtin_amdgcn_s_wait_asynccnt)
#define USE_ASYNC_LDS 1
#endif
#if __has_builtin(__builtin_amdgcn_tensor_load_to_lds) && \
    __has_builtin(__builtin_amdgcn_s_wait_tensorcnt)
#define USE_TDM 1
#endif
#endif
#ifndef USE_ASYNC_LDS
#define USE_ASYNC_LDS 0
#endif
#ifndef USE_TDM
#define USE_TDM 0
#endif

#if USE_ASYNC_LDS
__device__ __forceinline__ void async_g2l_b32(const float* g, float* l) {
  // Builtin params: (__device__ int*, __shared__ int*, imm offset, imm cpol).
  // LDS generic addresses carry the LDS byte offset in the low 32 bits.
  __builtin_amdgcn_global_load_async_to_lds_b32(
      (__attribute__((address_space(1))) int*)(uintptr_t)g,
      (__attribute__((address_space(3))) int*)(uint32_t)(uintptr_t)l,
      0, 0);
}
#endif

#if USE_TDM
typedef unsigned int tdm_v4u __attribute__((ext_vector_type(4)));
typedef int tdm_v8i __attribute__((ext_vector_type(8)));
typedef int tdm_v4i __attribute__((ext_vector_type(4)));

// 2D TDM descriptor: copy an IDIM x IDIM f32 tile (row stride IMW) from
// global memory into a contiguous LDS staging region.
// This toolchain exposes the 6-arg builtin:
//   (v4u group0, v8i group1, v4i group2, v4i group3, v8i extra, i32 cpol)
__device__ __forceinline__ void tdm_load_tile(const float* gsrc,
                                              uint32_t lds_byte_addr) {
  uint64_t ga = (uint64_t)(uintptr_t)gsrc;
  tdm_v4u g0;
  g0[0] = 0x1u;                                  // count=1 user descriptor
  g0[1] = lds_byte_addr;                         // lds_addr (bytes)
  g0[2] = (uint32_t)(ga & 0xFFFFFFFFu);          // global_addr[31:0]
  g0[3] = (uint32_t)((ga >> 32) & 0x1FFFFFFu)    // global_addr[56:32]
          | (2u << 30);                          // type = 2 ("image")
  tdm_v8i g1;
  g1[0] = (int)(2u << 16);                       // wg_mask=0, data_size=4B
  g1[1] = (int)(((uint32_t)IMW & 0xFFFFu) << 16);          // tensor_dim0 lo16
  g1[2] = (int)(((uint32_t)IMW >> 16) |
                (((uint32_t)IMH & 0xFFFFu) << 16));        // dim0 hi | dim1 lo
  g1[3] = (int)(((uint32_t)IMH >> 16) |
                ((uint32_t)IDIM << 16));                   // dim1 hi | tile_dim0
  g1[4] = (int)IDIM;                             // tile_dim1 (tile_dim2=0)
  g1[5] = (int)IMW;                              // tensor_dim0_stride[31:0]
  g1[6] = 0;                                     // stride hi / dim1_stride lo
  g1[7] = 0;                                     // dim1_stride hi (2D)
  tdm_v4i z4 = {0, 0, 0, 0};                     // groups 2/3 unused (<=2D)
  tdm_v8i z8 = {0, 0, 0, 0, 0, 0, 0, 0};
  __builtin_amdgcn_tensor_load_to_lds(g0, g1, z4, z4, z8, 0);
}
#endif

// ---------------------------------------------------------------------------
// Kernel 1: contour = mask * (1 - erode3(mask)); erode pads with 0.
// ---------------------------------------------------------------------------
__global__ __launch_bounds__(THREADS) void k_contour(
    const float* __restrict__ tgt, float* __restrict__ contour) {
  int idx = blockIdx.x * THREADS + threadIdx.x;
  if (idx >= NPIX) return;
  int x = idx % IMW;
  int y = (idx / IMW) % IMH;
  const float* img = tgt + (size_t)(idx / (IMW * IMH)) * IMH * IMW;
  float m = img[y * IMW + x];
  float e = m;
#pragma unroll
  for (int dy = -1; dy <= 1; ++dy)
#pragma unroll
    for (int dx = -1; dx <= 1; ++dx) {
      int nx = x + dx, ny = y + dy;
      float v = (nx >= 0 && nx < IMW && ny >= 0 && ny < IMH)
                    ? img[ny * IMW + nx] : 0.0f;
      e = fminf(e, v);
    }
  contour[idx] = (m > 0.5f && e < 0.5f) ? 1.0f : 0.0f;
}

// ---------------------------------------------------------------------------
// Kernel 2: boundary = dilate3(contour) (pads 0); d0 = boundary ? 0 : BIG.
// ---------------------------------------------------------------------------
__global__ __launch_bounds__(THREADS) void k_init(
    const float* __restrict__ contour, float* __restrict__ d0) {
  int idx = blockIdx.x * THREADS + threadIdx.x;
  if (idx >= NPIX) return;
  int x = idx % IMW;
  int y = (idx / IMW) % IMH;
  const float* img = contour + (size_t)(idx / (IMW * IMH)) * IMH * IMW;
  float b = 0.0f;
#pragma unroll
  for (int dy = -1; dy <= 1; ++dy)
#pragma unroll
    for (int dx = -1; dx <= 1; ++dx) {
      int nx = x + dx, ny = y + dy;
      if (nx >= 0 && nx < IMW && ny >= 0 && ny < IMH)
        b = fmaxf(b, img[ny * IMW + nx]);
    }
  d0[idx] = (b > 0.5f) ? 0.0f : BIGD;
}

// ---------------------------------------------------------------------------
// Kernel 3: one chamfer pass = RIT Jacobi min-plus iterations inside LDS.
// Guard ring held at BIG reproduces jnp.pad(..., BIG). Halo = RIT makes the
// written 64x64 center bit-identical to RIT full-image sweeps.
// ---------------------------------------------------------------------------
__global__ __launch_bounds__(THREADS) void k_chamfer(
    const float* __restrict__ src, float* __restrict__ dst) {
  __shared__ float buf[2][PCELLS];   // 53.8KB of the 320KB WGP LDS
  const int tid = threadIdx.x;
  const float* s = src + (size_t)blockIdx.z * IMH * IMW;
  float* d = dst + (size_t)blockIdx.z * IMH * IMW;
  const int gx0 = blockIdx.x * TILE - HALO;
  const int gy0 = blockIdx.y * TILE - HALO;

  // Per-thread cell coordinates (invariant across iterations).
  int px[CPT];
#pragma unroll
  for (int k = 0; k < CPT; ++k) {
    int i = tid + k * THREADS;
    px[k] = (i / IDIM + 1) * PDIM + (i % IDIM) + 1;
  }

  for (int i = tid; i < PCELLS; i += THREADS) {
    buf[0][i] = BIGD;
    buf[1][i] = BIGD;                 // dst-buffer guard ring stays BIG too
  }
  __syncthreads();

#if USE_TDM
  const bool interior = (gx0 >= 0) && (gy0 >= 0) &&
                        (gx0 + IDIM <= IMW) && (gy0 + IDIM <= IMH);
  if (interior) {
    // One DMA descriptor per tile, issued by wave 0 only (EXEC-independent);
    // staged contiguously into buf[1], then scattered into the ring layout.
    if (tid < 32) {
      tdm_load_tile(&s[gy0 * IMW + gx0], (uint32_t)(uintptr_t)&buf[1][0]);
      __builtin_amdgcn_s_wait_tensorcnt(0);
    }
    __syncthreads();
#pragma unroll
    for (int k = 0; k < CPT; ++k) {
      int i = tid + k * THREADS;
      buf[0][px[k]] = buf[1][i];
    }
    __syncthreads();
    for (int i = tid; i < PCELLS; i += THREADS) buf[1][i] = BIGD;
    __syncthreads();
  } else
#endif
  {
#pragma unroll
    for (int k = 0; k < CPT; ++k) {
      int i = tid + k * THREADS;
      int x = i % IDIM, y = i / IDIM;
      int gx = gx0 + x, gy = gy0 + y;
      if (gx >= 0 && gx < IMW && gy >= 0 && gy < IMH) {
#if USE_ASYNC_LDS
        async_g2l_b32(&s[gy * IMW + gx], &buf[0][px[k]]); // EXEC-masked async
#else
        buf[0][px[k]] = s[gy * IMW + gx];
#endif
      }
    }
#if USE_ASYNC_LDS
    __builtin_amdgcn_s_wait_asynccnt(0);
#endif
    __syncthreads();
  }

  for (int it = 0; it < RIT; ++it) {
    const float* sb = buf[it & 1];
    float* db = buf[(it & 1) ^ 1];
#pragma unroll
    for (int k = 0; k < CPT; ++k) {
      int p = px[k];
      float c = sb[p];
      c = fminf(c, sb[p - 1] + W_EDGE);
      c = fminf(c, sb[p + 1] + W_EDGE);
      c = fminf(c, sb[p - PDIM] + W_EDGE);
      c = fminf(c, sb[p + PDIM] + W_EDGE);
      c = fminf(c, sb[p - PDIM - 1] + W_DIAG);
      c = fminf(c, sb[p - PDIM + 1] + W_DIAG);
      c = fminf(c, sb[p + PDIM - 1] + W_DIAG);
      c = fminf(c, sb[p + PDIM + 1] + W_DIAG);
      db[p] = c;
    }
    __syncthreads();
  }

  // RIT even -> final iterate in buf[0]; store exact 64x64 center.
#pragma unroll
  for (int k = 0; k < OPT; ++k) {
    int i = tid + k * THREADS;
    int x = i % TILE, y = i / TILE;
    int p = (y + HALO + 1) * PDIM + (x + HALO + 1);
    d[(blockIdx.y * TILE + y) * IMW + (blockIdx.x * TILE + x)] = buf[0][p];
  }
}

// ---------------------------------------------------------------------------
// Kernel 4: weighted BCE partial sums (fixed order -> deterministic replays).
// ---------------------------------------------------------------------------
__global__ __launch_bounds__(THREADS) void k_loss(
    const float* __restrict__ lg, const float* __restrict__ tg,
    const float* __restrict__ dist, float* __restrict__ partials) {
  __shared__ float red[THREADS];
  const int tid = threadIdx.x;
  const size_t base = (size_t)blockIdx.x * (THREADS * PER_THREAD);
  float acc = 0.0f;
#pragma unroll 4
  for (int i = 0; i < PER_THREAD; ++i) {
    size_t idx = base + (size_t)i * THREADS + tid;
    if (i + 1 < PER_THREAD) {
      __builtin_prefetch(&lg[idx + THREADS], 0, 0);   // global_prefetch_b8
      __builtin_prefetch(&tg[idx + THREADS], 0, 0);
    }
    float x = lg[idx];
    float t = tg[idx];
    float dd = dist[idx];
    // sigmoid -> clip -> logit -> BCE-with-logits (exact reference chain)
    float p = 1.0f / (1.0f + expf(-x));
    p = fminf(fmaxf(p, 1e-7f), 1.0f - 1e-7f);
    float l = logf(p) - log1pf(-p);
    float bce = fmaxf(l, 0.0f) - l * t + log1pf(expf(-fabsf(l)));
    float w = expf(dd * (-1.0f / 3.0f)) + 0.1f;
    acc += bce * w;
  }
  red[tid] = acc;
  __syncthreads();
#pragma unroll
  for (int sft = THREADS / 2; sft > 0; sft >>= 1) {
    if (tid < sft) red[tid] += red[tid + sft];
    __syncthreads();
  }
  if (tid == 0) partials[blockIdx.x] = red[0];
}

// ---------------------------------------------------------------------------
// Kernel 5: deterministic final reduction -> mean.
// ---------------------------------------------------------------------------
__global__ __launch_bounds__(THREADS) void k_final(
    const float* __restrict__ partials, float* __restrict__ out) {
  __shared__ float red[THREADS];
  const int tid = threadIdx.x;
  red[tid] = partials[tid] + partials[tid + 256] +
             partials[tid + 512] + partials[tid + 768];
  __syncthreads();
#pragma unroll
  for (int sft = THREADS / 2; sft > 0; sft >>= 1) {
    if (tid < sft) red[tid] += red[tid + sft];
    __syncthreads();
  }
  if (tid == 0) out[0] = red[0] * (1.0f / (float)NPIX);
}

// ---------------------------------------------------------------------------
extern "C" void kernel_launch(void* const* d_in, const int* in_sizes, int n_in,
                              void* d_out, int out_size, void* d_ws,
                              size_t ws_size, hipStream_t stream) {
  (void)in_sizes; (void)n_in; (void)out_size; (void)ws_size;
  const float* inp = (const float*)d_in[0];   // logits  [16,1,512,512] f32
  const float* tgt = (const float*)d_in[1];   // targets [16,1,512,512] f32
  float* out = (float*)d_out;                 // scalar f32

  float* dA = (float*)d_ws;                   // ping
  float* dB = dA + NPIX;                      // pong (also holds contour)
  float* partials = dB + NPIX;                // 1024 floats

  const int nblk = NPIX / THREADS;
  k_contour<<<nblk, THREADS, 0, stream>>>(tgt, dB);
  k_init<<<nblk, THREADS, 0, stream>>>(dB, dA);

  dim3 grid(IMW / TILE, IMH / TILE, BATCH);   // 8 x 8 x 16 = 1024 blocks
  float* s = dA;
  float* d = dB;
  for (int p = 0; p < NPASS; ++p) {           // 8 passes x 8 LDS iters = 64
    k_chamfer<<<grid, THREADS, 0, stream>>>(s, d);
    float* t = s; s = d; d = t;
  }                                           // NPASS even -> result in dA (s)

  k_loss<<<LOSS_BLOCKS, THREADS, 0, stream>>>(inp, tgt, s, partials);
  k_final<<<1, THREADS, 0, stream>>>(partials, out);
}